// Loss_39170101740023
// MI455X (gfx1250) — compile-verified
//
#include <hip/hip_runtime.h>

#define BB   8
#define NP   4096      // fine / gt points per batch
#define KP   512       // coarse points per batch
#define EPSR 0.005f
#define SITERS 50

typedef float v2f __attribute__((ext_vector_type(2)));
typedef float v8f __attribute__((ext_vector_type(8)));

// ---------------------------------------------------------------- init
__global__ void init_kernel(unsigned* __restrict__ colmin) {
    int t = blockIdx.x * blockDim.x + threadIdx.x;
    if (t < BB * NP) colmin[t] = 0x7F800000u;   // +inf bits
}

// ---------------------------------------------------------------- FPS
// One block (256 threads) per batch. Each thread caches 16 strided points
// (registers only; no dynamic register indexing), runs 512 argmax rounds
// with first-index tie-break to match jnp.argmax.
__global__ void fps_kernel(const float* __restrict__ gt,
                           float* __restrict__ gt_ds) {
    const int b = blockIdx.x;
    const int t = threadIdx.x;                 // 256 threads
    const float* gx = gt + (size_t)b * 3 * NP;
    const float* gy = gx + NP;
    const float* gz = gy + NP;

    float lx[16], ly[16], lz[16], md[16];
#pragma unroll
    for (int s = 0; s < 16; ++s) {
        int i = t + s * 256;
        lx[s] = gx[i]; ly[s] = gy[i]; lz[s] = gz[i];
        md[s] = __builtin_inff();
    }

    __shared__ float sel[3];
    __shared__ float redv[256];
    __shared__ int   redi[256];

    int cur = 0;
    for (int k = 0; k < KP; ++k) {
        if (t == 0) {
            float px = gx[cur], py = gy[cur], pz = gz[cur];
            sel[0] = px; sel[1] = py; sel[2] = pz;
            float* o = gt_ds + ((size_t)b * KP + k) * 3;
            o[0] = px; o[1] = py; o[2] = pz;
        }
        __syncthreads();
        if (k == KP - 1) break;

        float sx = sel[0], sy = sel[1], sz = sel[2];
        float bestv = -1.0f; int besti = 0;
#pragma unroll
        for (int s = 0; s < 16; ++s) {
            float dx = lx[s] - sx, dy = ly[s] - sy, dz = lz[s] - sz;
            float d = dx * dx + dy * dy + dz * dz;
            float m = fminf(md[s], d);
            md[s] = m;
            if (m > bestv) { bestv = m; besti = t + s * 256; }  // ascending idx
        }
        redv[t] = bestv; redi[t] = besti;
        __syncthreads();
        for (int off = 128; off > 0; off >>= 1) {
            if (t < off) {
                float v2 = redv[t + off]; int i2 = redi[t + off];
                if (v2 > redv[t] || (v2 == redv[t] && i2 < redi[t])) {
                    redv[t] = v2; redi[t] = i2;
                }
            }
            __syncthreads();
        }
        cur = redi[0];
        __syncthreads();
    }
}

// ---------------------------------------------------------------- Sinkhorn
// One block (512 threads) per batch; points + duals live in LDS, C_ij
// recomputed on the fly (3 FMAs) instead of 1 MB/batch storage.
// logsumexp = max + log(sum exp(x-max)) -> matches jax.nn.logsumexp.
__global__ void sinkhorn_kernel(const float* __restrict__ coarse,
                                const float* __restrict__ gt_ds,
                                float* __restrict__ emd_c) {
    const int b = blockIdx.x;
    const int i = threadIdx.x;                 // 512 threads

    __shared__ float X0[KP], X1[KP], X2[KP], XX[KP];
    __shared__ float Y0[KP], Y1[KP], Y2[KP], YY[KP];
    __shared__ float U[KP], V[KP];

    {
        const float* cp = coarse + ((size_t)b * KP + i) * 3;
        float a0 = cp[0], a1 = cp[1], a2 = cp[2];
        X0[i] = a0; X1[i] = a1; X2[i] = a2; XX[i] = a0*a0 + a1*a1 + a2*a2;
        const float* gp = gt_ds + ((size_t)b * KP + i) * 3;
        float c0 = gp[0], c1 = gp[1], c2 = gp[2];
        Y0[i] = c0; Y1[i] = c1; Y2[i] = c2; YY[i] = c0*c0 + c1*c1 + c2*c2;
        U[i] = 0.0f; V[i] = 0.0f;
    }
    __syncthreads();

    const float inv_eps = 1.0f / EPSR;
    const float log_w = -logf((float)KP);
    const float x0 = X0[i], x1 = X1[i], x2 = X2[i], xxi = XX[i];
    const float y0 = Y0[i], y1 = Y1[i], y2 = Y2[i], yyi = YY[i];

    for (int it = 0; it < SITERS; ++it) {
        // ---- u_i = log_w - logsumexp_j( -C_ij/eps + v_j )
        float m = -__builtin_inff();
        for (int j = 0; j < KP; ++j) {
            float c = fmaxf(xxi + YY[j] - 2.0f * (x0*Y0[j] + x1*Y1[j] + x2*Y2[j]), 0.0f);
            m = fmaxf(m, V[j] - c * inv_eps);
        }
        float s = 0.0f;
        for (int j = 0; j < KP; ++j) {
            float c = fmaxf(xxi + YY[j] - 2.0f * (x0*Y0[j] + x1*Y1[j] + x2*Y2[j]), 0.0f);
            s += expf((V[j] - c * inv_eps) - m);
        }
        U[i] = log_w - (m + logf(s));
        __syncthreads();

        // ---- v_j = log_w - logsumexp_i( -C_ij/eps + u_i )   (thread i == col j)
        m = -__builtin_inff();
        for (int j = 0; j < KP; ++j) {
            float c = fmaxf(XX[j] + yyi - 2.0f * (X0[j]*y0 + X1[j]*y1 + X2[j]*y2), 0.0f);
            m = fmaxf(m, U[j] - c * inv_eps);
        }
        s = 0.0f;
        for (int j = 0; j < KP; ++j) {
            float c = fmaxf(XX[j] + yyi - 2.0f * (X0[j]*y0 + X1[j]*y1 + X2[j]*y2), 0.0f);
            s += expf((U[j] - c * inv_eps) - m);
        }
        V[i] = log_w - (m + logf(s));
        __syncthreads();
    }

    // dist_i = N * sum_j exp(u_i - C/eps + v_j) * C
    float ui = U[i];
    float s2 = 0.0f;
    for (int j = 0; j < KP; ++j) {
        float c = fmaxf(xxi + YY[j] - 2.0f * (x0*Y0[j] + x1*Y1[j] + x2*Y2[j]), 0.0f);
        s2 += expf(ui - c * inv_eps + V[j]) * c;
    }
    float dist = (float)KP * s2;
    emd_c[(size_t)b * KP + i] = sqrtf(fmaxf(dist, 1e-12f));
}

// ---------------------------------------------------------------- Chamfer (WMMA)
// 256 threads = 8 waves; each wave owns a 16-row x-tile and sweeps all 256
// y-tiles with V_WMMA_F32_16X16X4_F32 (A = -2*x padded to K=4, B = y).
// j-loop is unrolled x4: 4 independent WMMA tiles in flight so the
// WMMA->VALU RAW hazard window (CDNA5 ISA 7.12.1) is covered by the
// neighbouring tiles' epilogues, and y loads cluster into one clause.
// Row mins -> per-wave partial sums; column mins -> LDS uint atomicMin,
// then one global uint atomicMin pass per block (order-independent).
__global__ void chamfer_kernel(const float* __restrict__ fine,
                               const float* __restrict__ gt,
                               unsigned* __restrict__ colmin,
                               float* __restrict__ rowpart) {
    __shared__ unsigned cmin[NP];              // 16 KB
    const int tid = threadIdx.x;               // 256
    for (int j = tid; j < NP; j += 256) cmin[j] = 0x7F800000u;
    __syncthreads();

    const int b     = blockIdx.x >> 5;         // 32 blocks per batch
    const int strip = blockIdx.x & 31;
    const int wave  = tid >> 5;
    const int lane  = tid & 31;
    const int half  = lane >> 4;
    const int l     = lane & 15;
    const int i0    = (strip * 8 + wave) * 16;

    const float* xp = fine + (size_t)b * NP * 3;
    const float* g0 = gt + (size_t)b * 3 * NP;
    const float* g1 = g0 + NP;
    const float* g2 = g1 + NP;

    // A operand: lanes 0-15 row l k={0,1}; lanes 16-31 row l k={2,3(=0)}
    v2f a;
    {
        const float* p = xp + (size_t)(i0 + l) * 3;
        if (half == 0) { a.x = -2.0f * p[0]; a.y = -2.0f * p[1]; }
        else           { a.x = -2.0f * p[2]; a.y = 0.0f; }
    }
    // ||x||^2 for the 8 rows this lane's D registers cover (rows 8*half+r)
    float xxv[8];
#pragma unroll
    for (int r = 0; r < 8; ++r) {
        const float* p = xp + (size_t)(i0 + 8 * half + r) * 3;
        xxv[r] = p[0]*p[0] + p[1]*p[1] + p[2]*p[2];
    }
    float rowmin[8];
#pragma unroll
    for (int r = 0; r < 8; ++r) rowmin[r] = __builtin_inff();

    for (int j0 = 0; j0 < NP; j0 += 64) {      // 4 tiles per iteration
        float yx[4], yh[4], yz[4], yyn[4];
#pragma unroll
        for (int t4 = 0; t4 < 4; ++t4) {
            int n = j0 + t4 * 16 + l;
            yx[t4] = g0[n]; yh[t4] = g1[n]; yz[t4] = g2[n];
            yyn[t4] = yx[t4]*yx[t4] + yh[t4]*yh[t4] + yz[t4]*yz[t4];
        }
        v8f c4[4];
#pragma unroll
        for (int t4 = 0; t4 < 4; ++t4) {
            v2f bv;
            if (half == 0) { bv.x = yx[t4]; bv.y = yh[t4]; }  // k={0,1}
            else           { bv.x = yz[t4]; bv.y = 0.0f;  }   // k={2,3(=0)}
            v8f c = {};
            // (neg_a, A, neg_b, B, c_mod, C, reuse_a, reuse_b)
            c4[t4] = __builtin_amdgcn_wmma_f32_16x16x4_f32(
                         false, a, false, bv, (short)0, c, false, false);
        }
#pragma unroll
        for (int t4 = 0; t4 < 4; ++t4) {
            int n = j0 + t4 * 16 + l;
            float cm = __builtin_inff();
#pragma unroll
            for (int r = 0; r < 8; ++r) {
                float d = fmaxf(c4[t4][r] + xxv[r] + yyn[t4], 0.0f);
                rowmin[r] = fminf(rowmin[r], d);
                cm = fminf(cm, d);
            }
            cm = fminf(cm, __shfl_xor(cm, 16, 32));   // merge rows 0-7 / 8-15
            if (half == 0) atomicMin(&cmin[n], __float_as_uint(cm));
        }
    }

    // row-min across the 16 lanes of each half-group
#pragma unroll
    for (int off = 1; off < 16; off <<= 1) {
#pragma unroll
        for (int r = 0; r < 8; ++r)
            rowmin[r] = fminf(rowmin[r], __shfl_xor(rowmin[r], off, 32));
    }
    if (l == 0) {   // lanes 0 (rows 0-7) and 16 (rows 8-15) write partials
        float s = 0.0f;
#pragma unroll
        for (int r = 0; r < 8; ++r) s += rowmin[r];
        rowpart[(size_t)(blockIdx.x * 8 + wave) * 2 + half] = s;
    }
    __syncthreads();
    unsigned* gcol = colmin + (size_t)b * NP;
    for (int j = tid; j < NP; j += 256) atomicMin(&gcol[j], cmin[j]);
}

// ---------------------------------------------------------------- finalize
__global__ void finalize_kernel(const unsigned* __restrict__ colmin,
                                const float* __restrict__ rowpart,
                                const float* __restrict__ emd_c,
                                const float* __restrict__ alpha_p,
                                float* __restrict__ out) {
    __shared__ float red[256];
    const int t = threadIdx.x;

    float cs = 0.0f;
    for (int j = t; j < BB * NP; j += 256) cs += __uint_as_float(colmin[j]);
    float rs = 0.0f;
    for (int j = t; j < BB * 32 * 8 * 2; j += 256) rs += rowpart[j];
    float es = 0.0f;
    for (int j = t; j < BB * KP; j += 256) es += emd_c[j];

    red[t] = cs; __syncthreads();
    for (int off = 128; off > 0; off >>= 1) { if (t < off) red[t] += red[t + off]; __syncthreads(); }
    float colsum = red[0]; __syncthreads();
    red[t] = rs; __syncthreads();
    for (int off = 128; off > 0; off >>= 1) { if (t < off) red[t] += red[t + off]; __syncthreads(); }
    float rowsum = red[0]; __syncthreads();
    red[t] = es; __syncthreads();
    for (int off = 128; off > 0; off >>= 1) { if (t < off) red[t] += red[t + off]; __syncthreads(); }
    float emdsum = red[0];

    if (t == 0) {
        float loss_fine   = rowsum / (float)(BB * NP) + colsum / (float)(BB * NP);
        float loss_coarse = emdsum / (float)(BB * KP);
        float alpha = alpha_p[0];
        out[0] = loss_coarse + alpha * loss_fine;
        out[1] = loss_coarse;
        out[2] = loss_fine;
    }
}

// ---------------------------------------------------------------- launch
extern "C" void kernel_launch(void* const* d_in, const int* in_sizes, int n_in,
                              void* d_out, int out_size, void* d_ws, size_t ws_size,
                              hipStream_t stream) {
    (void)in_sizes; (void)n_in; (void)out_size; (void)ws_size;
    const float* coarse = (const float*)d_in[0];   // [8,512,3]
    const float* fine   = (const float*)d_in[1];   // [8,4096,3]
    const float* gt     = (const float*)d_in[2];   // [8,3,4096]
    const float* alpha  = (const float*)d_in[3];   // scalar
    float* out = (float*)d_out;                    // 3 floats

    float*    gt_ds   = (float*)d_ws;                       // 8*512*3
    unsigned* colmin  = (unsigned*)(gt_ds + BB * KP * 3);   // 8*4096
    float*    emd_c   = (float*)(colmin + BB * NP);         // 8*512
    float*    rowpart = emd_c + BB * KP;                    // 8*32*8*2 = 4096

    init_kernel<<<(BB * NP + 255) / 256, 256, 0, stream>>>(colmin);
    fps_kernel<<<BB, 256, 0, stream>>>(gt, gt_ds);
    sinkhorn_kernel<<<BB, KP, 0, stream>>>(coarse, gt_ds, emd_c);
    chamfer_kernel<<<BB * 32, 256, 0, stream>>>(fine, gt, colmin, rowpart);
    finalize_kernel<<<1, 256, 0, stream>>>(colmin, rowpart, emd_c, alpha, out);
}